// RadiusGraph_39539468927634
// MI455X (gfx1250) — compile-verified
//
#include <hip/hip_runtime.h>
#include <stdint.h>

typedef float v2f __attribute__((ext_vector_type(2)));
typedef float v3f __attribute__((ext_vector_type(3)));
typedef float v8f __attribute__((ext_vector_type(8)));
// reduced-alignment vec3 so a b96 store at 4-byte alignment is legal
typedef v3f uv3f __attribute__((aligned(4)));

#define TI 16
#define TJ 128
#define R_CUT2 25.0f   // R = 5.0 from reference

// ---------------------------------------------------------------------------
// LDS helpers: low 32 bits of a flat shared-aperture pointer == LDS byte offset
// (ISA 10.2: LDS aperture maps addr[31:0] directly to LDS_ADDR).
// ---------------------------------------------------------------------------
__device__ __forceinline__ uint32_t lds_off_u32(const void* p) {
  return (uint32_t)(uintptr_t)p;
}

// CDNA5 async global->LDS copy (ASYNCcnt path), 16B per lane, GV addressing.
__device__ __forceinline__ void async_copy_b128(uint32_t lds_byte_off, const void* gaddr) {
  asm volatile("global_load_async_to_lds_b128 %0, %1, off"
               :: "v"(lds_byte_off), "v"(gaddr)
               : "memory");
}

__device__ __forceinline__ void wait_async_all() {
  asm volatile("s_wait_asynccnt 0" ::: "memory");
}

__device__ __forceinline__ void store_disp_nt(float* p, float x, float y, float z) {
  v3f v = {x, y, z};
  __builtin_nontemporal_store(v, (uv3f*)p);
}

// ---------------------------------------------------------------------------
// Kernel A: per-batch (cell, inv(cell^T)) -> ws[b*18 + 0..8]=cell, +9..17=Minv
// ---------------------------------------------------------------------------
__global__ void rg_invert_cells(const float* __restrict__ cell,
                                float* __restrict__ ws, int nb) {
  int b = threadIdx.x;
  if (b >= nb) return;
  const float* c = cell + b * 9;
  // A = cell^T : A[r][k] = cell[k][r]
  float a00 = c[0], a01 = c[3], a02 = c[6];
  float a10 = c[1], a11 = c[4], a12 = c[7];
  float a20 = c[2], a21 = c[5], a22 = c[8];
  float c00 = a11 * a22 - a12 * a21;
  float c01 = a12 * a20 - a10 * a22;
  float c02 = a10 * a21 - a11 * a20;
  float det = a00 * c00 + a01 * c01 + a02 * c02;
  float id = 1.0f / det;
  float* o = ws + b * 18;
  #pragma unroll
  for (int k = 0; k < 9; ++k) o[k] = c[k];
  float* m = o + 9;
  m[0] = c00 * id;
  m[1] = (a02 * a21 - a01 * a22) * id;
  m[2] = (a01 * a12 - a02 * a11) * id;
  m[3] = c01 * id;
  m[4] = (a00 * a22 - a02 * a20) * id;
  m[5] = (a02 * a10 - a00 * a12) * id;
  m[6] = c02 * id;
  m[7] = (a01 * a20 - a00 * a21) * id;
  m[8] = (a00 * a11 - a01 * a10) * id;
}

// ---------------------------------------------------------------------------
// Per-pair wrap math given d and matrices; returns masked (w, fm) via refs.
// ---------------------------------------------------------------------------
__device__ __forceinline__ void wrap_pair(
    float dx, float dy, float dz,
    float mi0, float mi1, float mi2, float mi3, float mi4,
    float mi5, float mi6, float mi7, float mi8,
    float cm0, float cm1, float cm2, float cm3, float cm4,
    float cm5, float cm6, float cm7, float cm8,
    bool graph_ok, bool not_self,
    float& ox, float& oy, float& oz, float& fm) {
  // s = inv(cell^T) * d
  float sx = fmaf(mi0, dx, fmaf(mi1, dy, mi2 * dz));
  float sy = fmaf(mi3, dx, fmaf(mi4, dy, mi5 * dz));
  float sz = fmaf(mi6, dx, fmaf(mi7, dy, mi8 * dz));
  // round half-to-even (matches jnp.round under default RNE)
  float rx = rintf(sx), ry = rintf(sy), rz = rintf(sz);
  // w = d - cell*r, folded as negated-FMA chains
  float wx = fmaf(-cm0, rx, fmaf(-cm1, ry, fmaf(-cm2, rz, dx)));
  float wy = fmaf(-cm3, rx, fmaf(-cm4, ry, fmaf(-cm5, rz, dy)));
  float wz = fmaf(-cm6, rx, fmaf(-cm7, ry, fmaf(-cm8, rz, dz)));
  float d2 = fmaf(wx, wx, fmaf(wy, wy, wz * wz));
  bool m = graph_ok && not_self && (d2 < R_CUT2);
  fm = m ? 1.0f : 0.0f;
  ox = m ? wx : 0.0f;
  oy = m ? wy : 0.0f;
  oz = m ? wz : 0.0f;
}

// ---------------------------------------------------------------------------
// Kernel B: block = 16(i) x 128(j), 8 waves, one 16x16 WMMA tile per wave.
// d_c tile via V_WMMA_F32_16X16X4_F32 rank-2 trick; wrap math on VALU;
// b96+b32 non-temporal streaming stores.
// ---------------------------------------------------------------------------
__global__ __launch_bounds__(256) void rg_pairs(
    const float* __restrict__ pos, const int* __restrict__ batch,
    const float* __restrict__ mats, float* __restrict__ out, int n) {

  __shared__ __align__(16) float s_posj[TJ * 3];   // 1536 B
  __shared__ __align__(16) float s_posi[TI * 3];   //  192 B
  __shared__ __align__(16) int   s_batchi[TI];     //   64 B
  __shared__ __align__(16) float s_mats[8 * 18];   //  576 B

  const int tid = threadIdx.x;
  const int j0 = blockIdx.x * TJ;
  const int i0 = blockIdx.y * TI;

  // ---- async stage of all block-shared inputs (single ASYNCcnt burst) ----
  if (tid < 96) {                                   // 96 * 16B = posj tile
    async_copy_b128(lds_off_u32(&s_posj[tid * 4]),
                    pos + (size_t)j0 * 3 + (size_t)tid * 4);
  } else if (tid < 108) {                           // 12 * 16B = posi tile
    int t = tid - 96;
    async_copy_b128(lds_off_u32(&s_posi[t * 4]),
                    pos + (size_t)i0 * 3 + (size_t)t * 4);
  } else if (tid < 112) {                           // 4 * 16B = batch rows
    int t = tid - 108;
    async_copy_b128(lds_off_u32(&s_batchi[t * 4]), batch + i0 + t * 4);
  } else if (tid < 148) {                           // 36 * 16B = 8x(cell,Minv)
    int t = tid - 112;
    async_copy_b128(lds_off_u32(&s_mats[t * 4]), mats + t * 4);
  }
  wait_async_all();
  __syncthreads();

  const int wv   = tid >> 5;        // wave id 0..7 -> j sub-tile
  const int lane = tid & 31;
  const int col  = lane & 15;       // = WMMA N index (j) and A row (i)
  const bool hi  = lane >= 16;      // K-half for A/B, +8 row offset for C
  const int jl   = wv * 16 + col;   // j within block tile
  const int j    = j0 + jl;

  const int   bj  = batch[j];
  const float pjx = s_posj[jl * 3 + 0];
  const float pjy = s_posj[jl * 3 + 1];
  const float pjz = s_posj[jl * 3 + 2];

  // A (16x4, rows = i-tile): lanes<16 hold K0,K1 = (pos_i[c], 1); lanes>=16 K2,K3 = 0.
  // B (4x16, cols = j):      V0 lanes<16 = row K0 = 1; lanes>=16 = row K1 = -pos_j[c]; V1 = 0.
  // => C[i][j] = pos_i[c] - pos_j[c], exact in f32.
  v8f zero = {};
  v2f ax = {hi ? 0.0f : s_posi[col * 3 + 0], hi ? 0.0f : 1.0f};
  v2f ay = {hi ? 0.0f : s_posi[col * 3 + 1], hi ? 0.0f : 1.0f};
  v2f az = {hi ? 0.0f : s_posi[col * 3 + 2], hi ? 0.0f : 1.0f};
  v2f bx = {hi ? -pjx : 1.0f, 0.0f};
  v2f by = {hi ? -pjy : 1.0f, 0.0f};
  v2f bz = {hi ? -pjz : 1.0f, 0.0f};

  v8f dxv = __builtin_amdgcn_wmma_f32_16x16x4_f32(false, ax, false, bx, (short)0, zero, false, false);
  v8f dyv = __builtin_amdgcn_wmma_f32_16x16x4_f32(false, ay, false, by, (short)0, zero, false, false);
  v8f dzv = __builtin_amdgcn_wmma_f32_16x16x4_f32(false, az, false, bz, (short)0, zero, false, false);

  float* __restrict__ maskout = out + (size_t)3 * n * n;

  // C-tile row base for this lane: i = i0 + v + 8*hi, v = 0..7
  const int ib = i0 + (hi ? 8 : 0);
  float* pd = out + ((size_t)ib * n + j) * 3;   // advances by 3n per v
  float* pm = maskout + ((size_t)ib * n + j);   // advances by n per v
  const size_t dstep = (size_t)n * 3;

  // Tile-uniform batch id? (batch is sorted: true for ~97% of i-tiles)
  const int b0  = __builtin_amdgcn_readfirstlane(s_batchi[0]);
  const int b15 = __builtin_amdgcn_readfirstlane(s_batchi[15]);

  if (b0 == b15) {
    // -------- fast path: one graph for the whole tile; matrices in SGPRs ---
    const float* g = mats + b0 * 18;             // uniform pointer -> s_load
    float cm0 = g[0], cm1 = g[1], cm2 = g[2];
    float cm3 = g[3], cm4 = g[4], cm5 = g[5];
    float cm6 = g[6], cm7 = g[7], cm8 = g[8];
    float mi0 = g[9], mi1 = g[10], mi2 = g[11];
    float mi3 = g[12], mi4 = g[13], mi5 = g[14];
    float mi6 = g[15], mi7 = g[16], mi8 = g[17];
    const bool graph_ok = (bj == b0);            // hoisted out of v-loop

    #pragma unroll
    for (int v = 0; v < 8; ++v) {
      float ox, oy, oz, fm;
      wrap_pair(dxv[v], dyv[v], dzv[v],
                mi0, mi1, mi2, mi3, mi4, mi5, mi6, mi7, mi8,
                cm0, cm1, cm2, cm3, cm4, cm5, cm6, cm7, cm8,
                graph_ok, (ib + v) != j, ox, oy, oz, fm);
      store_disp_nt(pd, ox, oy, oz);
      __builtin_nontemporal_store(fm, pm);
      pd += dstep;
      pm += n;
    }
  } else {
    // -------- general path: per-row batch ids, matrices from LDS ----------
    #pragma unroll
    for (int v = 0; v < 8; ++v) {
      const int ir = v + (hi ? 8 : 0);
      const int bi = s_batchi[ir];
      const float* Cm = &s_mats[bi * 18];
      const float* Mi = Cm + 9;
      float ox, oy, oz, fm;
      wrap_pair(dxv[v], dyv[v], dzv[v],
                Mi[0], Mi[1], Mi[2], Mi[3], Mi[4], Mi[5], Mi[6], Mi[7], Mi[8],
                Cm[0], Cm[1], Cm[2], Cm[3], Cm[4], Cm[5], Cm[6], Cm[7], Cm[8],
                bi == bj, (ib + v) != j, ox, oy, oz, fm);
      store_disp_nt(pd, ox, oy, oz);
      __builtin_nontemporal_store(fm, pm);
      pd += dstep;
      pm += n;
    }
  }
}

// ---------------------------------------------------------------------------
extern "C" void kernel_launch(void* const* d_in, const int* in_sizes, int n_in,
                              void* d_out, int out_size, void* d_ws, size_t ws_size,
                              hipStream_t stream) {
  const float* pos   = (const float*)d_in[0];
  const float* cell  = (const float*)d_in[1];
  const int*   batch = (const int*)d_in[2];
  const int n  = in_sizes[0] / 3;   // 4096
  const int nb = in_sizes[1] / 9;   // 8

  float* ws = (float*)d_ws;         // 8 * 18 floats = 576 B

  rg_invert_cells<<<1, 64, 0, stream>>>(cell, ws, nb);

  dim3 grid(n / TJ, n / TI);        // (32, 256)
  rg_pairs<<<grid, 256, 0, stream>>>(pos, batch, ws, (float*)d_out, n);
}